// EndToEndMemoryNetwork_21500606284412
// MI455X (gfx1250) — compile-verified
//
#include <hip/hip_runtime.h>
#include <hip/hip_bf16.h>
#include <cstddef>

// ---------------------------------------------------------------------------
// End-to-End Memory Network, MI455X (gfx1250, wave32).
// Dominant cost: embedding gather-sum (25.6MB table -> L2 resident) and the
// 52MB ctx tensor (kept in ws, re-read from L2 per hop). The two dense GEMMs
// use V_WMMA_F32_16X16X4_F32 (f32-exact matrix path).
// ---------------------------------------------------------------------------

namespace {
constexpr int kB    = 2048;
constexpr int kS    = 50;
constexpr int kL    = 32;
constexpr int kQ    = 32;
constexpr int kD    = 128;
constexpr int kHops = 3;
}

typedef __attribute__((ext_vector_type(2))) float v2f;
typedef __attribute__((ext_vector_type(8))) float v8f;

// ---------------------------------------------------------------------------
// ctx[b,s,:] = sum_l input_emb[context[b,s,l],:] + pos_enc[s,:]
// one 128-thread block per (b,s); thread = d
// ---------------------------------------------------------------------------
__global__ void ctx_encode_kernel(const int* __restrict__ context,
                                  const float* __restrict__ emb,
                                  const float* __restrict__ pos,
                                  float* __restrict__ ctx) {
  const int bs = blockIdx.x;            // 0 .. B*S-1
  const int d  = threadIdx.x;           // 0 .. 127
  const int s  = bs % kS;
  const int* __restrict__ row = context + (size_t)bs * kL;
  float acc = pos[(size_t)s * kD + d];
  #pragma unroll 8
  for (int l = 0; l < kL; ++l) {
    acc += emb[(size_t)row[l] * kD + d];
  }
  ctx[(size_t)bs * kD + d] = acc;
}

// ---------------------------------------------------------------------------
// query[b,:] = sum_q question_emb[question[b,q],:]
// ---------------------------------------------------------------------------
__global__ void query_encode_kernel(const int* __restrict__ question,
                                    const float* __restrict__ emb,
                                    float* __restrict__ query) {
  const int b = blockIdx.x;
  const int d = threadIdx.x;
  const int* __restrict__ row = question + (size_t)b * kQ;
  float acc = 0.0f;
  #pragma unroll 8
  for (int q = 0; q < kQ; ++q) {
    acc += emb[(size_t)row[q] * kD + d];
  }
  query[(size_t)b * kD + d] = acc;
}

// ---------------------------------------------------------------------------
// pq = query @ W^T + bias  via V_WMMA_F32_16X16X4_F32.
// One wave (32 threads) computes a 16(batch) x 16(dout) tile; K=128 -> 32 WMMAs.
// A layout (16x4 f32): lane%16 = M, K = (lane/16)*2 + vgpr  -> contiguous f32x2.
// B layout (4x16 f32): lane%16 = N, K = (lane/16)*2 + vgpr; B[k][n] = W[n][k],
// so B also loads as contiguous f32x2 from the row-major W[n,:].
// ---------------------------------------------------------------------------
__global__ void hop_project_kernel(const float* __restrict__ query,
                                   const float* __restrict__ W,      // [D,D] (out,in)
                                   const float* __restrict__ bias,   // [D]
                                   float* __restrict__ pq) {
  const int m0    = blockIdx.x * 16;          // batch tile
  const int n0    = blockIdx.y * 16;          // out-dim tile
  const int lane  = threadIdx.x;              // 0..31 (one wave per block)
  const int r16   = lane & 15;
  const int khalf = (lane >> 4) << 1;         // 0 or 2

  const float* __restrict__ aRow = query + (size_t)(m0 + r16) * kD;
  const float* __restrict__ bRow = W     + (size_t)(n0 + r16) * kD;

  v8f c = {};
  #pragma unroll
  for (int kk = 0; kk < kD; kk += 4) {
    v2f a = *(const v2f*)(aRow + kk + khalf);
    v2f b = *(const v2f*)(bRow + kk + khalf);
    c = __builtin_amdgcn_wmma_f32_16x16x4_f32(
        /*neg_a=*/false, a, /*neg_b=*/false, b,
        /*c_mod=*/(short)0, c, /*reuse_a=*/false, /*reuse_b=*/false);
  }

  const int   nCol  = n0 + r16;
  const float bv    = bias[nCol];
  const int   mBase = m0 + ((lane >> 4) ? 8 : 0);
  #pragma unroll
  for (int r = 0; r < 8; ++r) {
    pq[(size_t)(mBase + r) * kD + nCol] = c[r] + bv;
  }
}

// ---------------------------------------------------------------------------
// Per-batch attention hop: scores = ctx[b] @ pq[b]; softmax; retrieved =
// attn^T @ ctx[b]; query[b] += retrieved; attn -> d_out.
// One 128-thread (4-wave) block per batch element; ctx tile staged in LDS.
// ---------------------------------------------------------------------------
__global__ void hop_attend_kernel(const float* __restrict__ ctx,
                                  const float* __restrict__ pq,
                                  float* __restrict__ query,
                                  float* __restrict__ attn_out) {
  const int b    = blockIdx.x;
  const int t    = threadIdx.x;   // 0..127
  const int wave = t >> 5;
  const int lane = t & 31;

  __shared__ float lds_ctx[kS * kD];   // 25.6 KB
  __shared__ float lds_pq[kD];
  __shared__ float lds_sc[64];

  const float* __restrict__ cb = ctx + (size_t)b * kS * kD;
  for (int i = t; i < kS * kD; i += 128) lds_ctx[i] = cb[i];
  lds_pq[t] = pq[(size_t)b * kD + t];
  __syncthreads();

  // scores: each wave handles s = wave, wave+4, ... ; 128-wide dot per s
  for (int s = wave; s < kS; s += 4) {
    float p = 0.0f;
    #pragma unroll
    for (int i = 0; i < 4; ++i) {
      const int d = lane + i * 32;
      p += lds_pq[d] * lds_ctx[s * kD + d];
    }
    #pragma unroll
    for (int off = 16; off > 0; off >>= 1) p += __shfl_xor(p, off, 32);
    if (lane == 0) lds_sc[s] = p;
  }
  __syncthreads();

  // softmax over the 50 scores by wave 0 (lane covers s=lane and s=lane+32)
  if (wave == 0) {
    const bool v0 = lane < kS;
    const bool v1 = (lane + 32) < kS;
    float s0 = v0 ? lds_sc[lane] : -1e30f;
    float s1 = v1 ? lds_sc[lane + 32] : -1e30f;
    float m = fmaxf(s0, s1);
    #pragma unroll
    for (int off = 16; off > 0; off >>= 1) m = fmaxf(m, __shfl_xor(m, off, 32));
    const float e0 = v0 ? __expf(s0 - m) : 0.0f;
    const float e1 = v1 ? __expf(s1 - m) : 0.0f;
    float sum = e0 + e1;
    #pragma unroll
    for (int off = 16; off > 0; off >>= 1) sum += __shfl_xor(sum, off, 32);
    const float inv = 1.0f / sum;
    if (v0) { lds_sc[lane]      = e0 * inv; attn_out[(size_t)b * kS + lane]      = e0 * inv; }
    if (v1) { lds_sc[lane + 32] = e1 * inv; attn_out[(size_t)b * kS + lane + 32] = e1 * inv; }
  }
  __syncthreads();

  // retrieved[d] = sum_s attn[s] * ctx[b,s,d]; query update in-place
  float acc = 0.0f;
  #pragma unroll 10
  for (int s = 0; s < kS; ++s) acc += lds_sc[s] * lds_ctx[s * kD + t];
  query[(size_t)b * kD + t] += acc;
}

// ---------------------------------------------------------------------------
// logits = query @ out_W^T + out_b   ([2048,128] x [128,50], WMMA f32,
// N padded to 64 via 4 tiles; rows clamped for n>=50, stores masked)
// ---------------------------------------------------------------------------
__global__ void out_project_kernel(const float* __restrict__ query,
                                   const float* __restrict__ outW,   // [S,D]
                                   const float* __restrict__ outb,   // [S]
                                   float* __restrict__ logits) {     // [B,S]
  const int m0    = blockIdx.x * 16;
  const int n0    = blockIdx.y * 16;
  const int lane  = threadIdx.x;
  const int r16   = lane & 15;
  const int khalf = (lane >> 4) << 1;

  const int  nIdx = n0 + r16;
  const int  nCl  = (nIdx < kS) ? nIdx : 0;   // clamped row; garbage cols never stored

  const float* __restrict__ aRow = query + (size_t)(m0 + r16) * kD;
  const float* __restrict__ bRow = outW  + (size_t)nCl * kD;

  v8f c = {};
  #pragma unroll
  for (int kk = 0; kk < kD; kk += 4) {
    v2f a = *(const v2f*)(aRow + kk + khalf);
    v2f b = *(const v2f*)(bRow + kk + khalf);
    c = __builtin_amdgcn_wmma_f32_16x16x4_f32(
        false, a, false, b, (short)0, c, false, false);
  }

  if (nIdx < kS) {
    const float bv    = outb[nIdx];
    const int   mBase = m0 + ((lane >> 4) ? 8 : 0);
    #pragma unroll
    for (int r = 0; r < 8; ++r) {
      logits[(size_t)(mBase + r) * kS + nIdx] = c[r] + bv;
    }
  }
}

// ---------------------------------------------------------------------------
extern "C" void kernel_launch(void* const* d_in, const int* in_sizes, int n_in,
                              void* d_out, int out_size, void* d_ws, size_t ws_size,
                              hipStream_t stream) {
  const int*   context      = (const int*)  d_in[0];   // [B,S,L]
  const int*   question     = (const int*)  d_in[1];   // [B,Q]
  const float* input_emb    = (const float*)d_in[2];   // [VOCAB,D]
  const float* question_emb = (const float*)d_in[3];   // [VOCAB,D]
  const float* pos_enc      = (const float*)d_in[4];   // [S,D]
  const float* hop_W        = (const float*)d_in[5];   // [HOPS,D,D]
  const float* hop_b        = (const float*)d_in[6];   // [HOPS,D]
  const float* out_W        = (const float*)d_in[7];   // [S,D]
  const float* out_b        = (const float*)d_in[8];   // [S]

  float* out = (float*)d_out;                // logits [B,S] then attn [HOPS,B,S]
  float* ws  = (float*)d_ws;

  float* ctx   = ws;                                     // B*S*D floats (52.4 MB)
  float* query = ctx   + (size_t)kB * kS * kD;           // B*D
  float* pq    = query + (size_t)kB * kD;                // B*D

  ctx_encode_kernel<<<kB * kS, kD, 0, stream>>>(context, input_emb, pos_enc, ctx);
  query_encode_kernel<<<kB, kD, 0, stream>>>(question, question_emb, query);

  for (int h = 0; h < kHops; ++h) {
    hop_project_kernel<<<dim3(kB / 16, kD / 16), 32, 0, stream>>>(
        query, hop_W + (size_t)h * kD * kD, hop_b + (size_t)h * kD, pq);
    hop_attend_kernel<<<kB, kD, 0, stream>>>(
        ctx, pq, query, out + (size_t)kB * kS + (size_t)h * kB * kS);
  }

  out_project_kernel<<<dim3(kB / 16, 4), 32, 0, stream>>>(query, out_W, out_b, out);
}